// IndRNN_67877663146291
// MI455X (gfx1250) — compile-verified
//
#include <hip/hip_runtime.h>

// IndRNN scan: h_t = relu(x_t + w * h_{t-1}),  x: (T,B,H) f32, out: (T,B,H) f32
// T=2048, B=32, H=1024. Pure streaming problem: 512 MB traffic -> ~22 us floor
// at 23.3 TB/s. One thread per (b,h) channel; async global->LDS ring buffer
// (GLOBAL_LOAD_ASYNC_TO_LDS_B32, tracked by ASYNCcnt) keeps ~48 copies in
// flight per wave without burning VGPRs; consume via ds_load, store NT.

#define T_STEPS   2048
#define B_DIM     32
#define H_DIM     1024
#define N_CH      (B_DIM * H_DIM)     // 32768 independent recurrences
#define BLOCK     256
#define GRID      (N_CH / BLOCK)      // 128 blocks = 1024 wave32s
#define RING      64                  // LDS ring slots (power of 2)
#define DEPTH     48                  // outstanding async copies per wave (< 63 cap)
#define UNROLL    4                   // steps consumed per s_wait_asynccnt
// (T_STEPS - DEPTH) % UNROLL == 0 and DEPTH % UNROLL == 0 by construction.

#if defined(__gfx1250__) && __has_builtin(__builtin_amdgcn_global_load_async_to_lds_b32)
#define HAVE_ASYNC 1
#else
#define HAVE_ASYNC 0
#endif

// Builtin signature (from hipcc diagnostic): params are int-typed AS pointers.
typedef __attribute__((address_space(1))) int   gbl_i32_t;
typedef __attribute__((address_space(3))) int   lds_i32_t;
typedef __attribute__((address_space(3))) float lds_f32_t;

// One 4-byte element per lane: a wave32 issue moves a contiguous 128B slab.
__device__ __forceinline__ void async_copy_b32(const float* gsrc, float* lds_dst) {
#if HAVE_ASYNC
  // Integer-routed addrspace casts: generic LDS pointer's low 32 bits are the
  // LDS byte offset (generic->local addrspacecast == trunc on AMDGPU).
  __builtin_amdgcn_global_load_async_to_lds_b32(
      (gbl_i32_t*)(unsigned long long)gsrc,
      (lds_i32_t*)(unsigned)(unsigned long long)lds_dst,
      /*imm offset*/ 0, /*cpol*/ 0);
#else
  *(volatile float*)lds_dst = *gsrc;   // functional fallback (synchronous)
#endif
}

// Read ring through an explicit as3 pointer -> ds_load_b32 (not flat_load).
// volatile: the async engine's LDS writes are invisible to the compiler.
__device__ __forceinline__ float lds_read_f32(const float* p) {
  const volatile lds_f32_t* lp =
      (const volatile lds_f32_t*)(unsigned)(unsigned long long)p;
  return *lp;
}

#if HAVE_ASYNC
#if __has_builtin(__builtin_amdgcn_s_wait_asynccnt)
#define WAIT_ASYNCCNT(n)                                   \
  do {                                                     \
    __builtin_amdgcn_s_wait_asynccnt(n);                   \
    asm volatile("" ::: "memory"); /* compiler fence */    \
  } while (0)
#else
#define WAIT_ASYNCCNT(n) asm volatile("s_wait_asynccnt %0" ::"i"(n) : "memory")
#endif
#else
#define WAIT_ASYNCCNT(n) do { } while (0)
#endif

__global__ __launch_bounds__(BLOCK) void indrnn_scan_kernel(
    const float* __restrict__ x,    // (T, B, H)
    const float* __restrict__ h0,   // (B, H)
    const float* __restrict__ whh,  // (H)
    float* __restrict__ out)        // (T, B, H)
{
  // 64 slots x 256 lanes x 4B = 64 KB ring per block (WGP has 320 KB LDS).
  __shared__ float ring[RING][BLOCK];

  const int tid = threadIdx.x;
  const int gid = blockIdx.x * BLOCK + tid;        // channel in [0, N_CH)

  const float w = whh[gid & (H_DIM - 1)];
  float h = h0[gid];

  const float* xp = x + gid;                       // stride N_CH per time step
  float*       op = out + gid;

  // ---- Prologue: prime the async pipeline DEPTH deep (loads t = 0..DEPTH-1).
#pragma unroll 1
  for (int i = 0; i < DEPTH; ++i) {
    async_copy_b32(xp + (long long)i * N_CH, &ring[i & (RING - 1)][tid]);
  }

  // ---- Steady state, UNROLL steps per wait:
  // At group start t: issued = t + DEPTH; wait ASYNCcnt <= DEPTH-UNROLL
  // guarantees completed >= t+UNROLL (in-order), i.e. slots t..t+3 resident.
  // Group slots (t&63)..(t&63)+3 never wrap (t % 4 == 0, RING = 64).
#pragma unroll 1
  for (int t = 0; t < T_STEPS - DEPTH; t += UNROLL) {
    WAIT_ASYNCCNT(DEPTH - UNROLL);

    float xv[UNROLL];
#pragma unroll
    for (int k = 0; k < UNROLL; ++k)
      xv[k] = lds_read_f32(&ring[(t + k) & (RING - 1)][tid]);

#pragma unroll
    for (int k = 0; k < UNROLL; ++k) {
      h = fmaxf(fmaf(w, h, xv[k]), 0.0f);          // relu(x + w*h)
      __builtin_nontemporal_store(h, op);          // 256MB stream: bypass L2 reuse
      op += N_CH;
    }

    asm volatile("" ::: "memory");  // reads retired (values consumed) before refills
    // Refill slots (t+DEPTH+k)&63: read 4 groups ago, long since retired.
#pragma unroll
    for (int k = 0; k < UNROLL; ++k)
      async_copy_b32(xp + (long long)(t + DEPTH + k) * N_CH,
                     &ring[(t + DEPTH + k) & (RING - 1)][tid]);
  }

  // ---- Tail: everything issued; drain the last DEPTH slots.
  WAIT_ASYNCCNT(0);
#pragma unroll 1
  for (int t = T_STEPS - DEPTH; t < T_STEPS; t += UNROLL) {
#pragma unroll
    for (int k = 0; k < UNROLL; ++k) {
      float xv = lds_read_f32(&ring[(t + k) & (RING - 1)][tid]);
      h = fmaxf(fmaf(w, h, xv), 0.0f);
      __builtin_nontemporal_store(h, op);
      op += N_CH;
    }
  }
}

extern "C" void kernel_launch(void* const* d_in, const int* in_sizes, int n_in,
                              void* d_out, int out_size, void* d_ws, size_t ws_size,
                              hipStream_t stream) {
  // setup_inputs() order: t_input (T,B,H) f32, h0 (B,H) f32, weight_hh (H) f32
  const float* x   = (const float*)d_in[0];
  const float* h0  = (const float*)d_in[1];
  const float* whh = (const float*)d_in[2];
  float* out = (float*)d_out;
  (void)in_sizes; (void)n_in; (void)out_size; (void)d_ws; (void)ws_size;

  indrnn_scan_kernel<<<GRID, BLOCK, 0, stream>>>(x, h0, whh, out);
}